// TGNN_5531917877811
// MI455X (gfx1250) — compile-verified
//
#include <hip/hip_runtime.h>

#define N_NODES 200000
#define N_EDGES 6400000

typedef float v2f __attribute__((ext_vector_type(2)));
typedef float v8f __attribute__((ext_vector_type(8)));

// ---------------------------------------------------------------------------
// ws float layout (needs (10*N + 130) floats ~ 8.0 MB):
//   [0,   N)   cnt          (zeroed each call)
//   [N,  4N)   sum_x  [N,3] (zeroed each call)
//   [4N, 6N)   sum_p  [N,2] (zeroed each call)
//   [6N, 8N)   p      [N,2]
//   [8N,10N)   q      [N,2]
//   [10N, +64) A2 = W2l@Wlin  [32,2]
//   [+64, +128) B2 = W2r@Wlin [32,2]
//   [+128,+130) c2 = b2@Wlin + blin [2]
// ---------------------------------------------------------------------------

__global__ void sage_zero_kernel(float* buf, int count) {
    int i = blockIdx.x * blockDim.x + threadIdx.x;
    if (i < count) buf[i] = 0.0f;
}

// Fold the linear head into layer-2 weights: A2 = W2l@Wlin, B2 = W2r@Wlin,
// c2 = b2@Wlin + blin.  W2l/W2r are [32,16] row-major, Wlin is [16,2].
__global__ void sage_fold_kernel(const float* __restrict__ W2l,
                                 const float* __restrict__ W2r,
                                 const float* __restrict__ Wlin,
                                 const float* __restrict__ b2,
                                 const float* __restrict__ blin,
                                 float* __restrict__ A2,
                                 float* __restrict__ B2,
                                 float* __restrict__ c2) {
    int t = threadIdx.x;                       // one block of 128 threads
    if (t < 64) {
        int n = t >> 1, j = t & 1;
        float sa = 0.0f, sb = 0.0f;
#pragma unroll
        for (int k = 0; k < 16; ++k) {
            float w = Wlin[k * 2 + j];
            sa = fmaf(W2l[n * 16 + k], w, sa);
            sb = fmaf(W2r[n * 16 + k], w, sb);
        }
        A2[t] = sa;
        B2[t] = sb;
    } else if (t < 66) {
        int j = t - 64;
        float s = blin[j];
#pragma unroll
        for (int k = 0; k < 16; ++k) s = fmaf(b2[k], Wlin[k * 2 + j], s);
        c2[j] = s;
    }
}

// Edge pass 1: per-dst degree count + sum of x[src] (3 floats).
// Node arrays (cnt 0.8MB, sum_x 2.4MB, x 2.4MB) are L2-resident on MI455X.
__global__ void sage_edge1_kernel(const int* __restrict__ ei,
                                  const float* __restrict__ x,
                                  float* __restrict__ cnt,
                                  float* __restrict__ sumx) {
    int e = blockIdx.x * blockDim.x + threadIdx.x;
    if (e >= N_EDGES) return;
    int src = ei[e];
    int dst = ei[N_EDGES + e];
    atomicAdd(&cnt[dst], 1.0f);
    atomicAdd(&sumx[dst * 3 + 0], x[src * 3 + 0]);
    atomicAdd(&sumx[dst * 3 + 1], x[src * 3 + 1]);
    atomicAdd(&sumx[dst * 3 + 2], x[src * 3 + 2]);
}

// Node kernel: per wave32, a 16-node tile.
//   A (16x8 = [mean | x], split into two K=4 WMMA chunks, K=3/7 zero-padded)
//   B ( 8x32 = [W1l ; W1r], two 16-column tiles)
//   D = relu(A@B + b1) via V_WMMA_F32_16X16X4_F32, bias preloaded in C.
// Then spill D through LDS (C-layout -> per-node rows) and project 32->2
// twice (A2 for the aggregated path, B2 for the root path).
__global__ __launch_bounds__(256) void sage_node_wmma_kernel(
        const float* __restrict__ x,
        const float* __restrict__ cnt,
        const float* __restrict__ sumx,
        const float* __restrict__ W1l,   // [3,32] row-major
        const float* __restrict__ W1r,   // [3,32] row-major
        const float* __restrict__ b1,    // [32]
        const float* __restrict__ A2,    // [32,2]
        const float* __restrict__ B2,    // [32,2]
        float* __restrict__ p,           // [N,2]
        float* __restrict__ q) {         // [N,2]
    __shared__ float h1s[8 * 16 * 33];   // 8 waves x 16 nodes x (32+1 pad)

    const int lane = threadIdx.x & 31;
    const int wave = threadIdx.x >> 5;
    const int row  = lane & 15;          // node-in-tile / N-column / B-column
    const int hi   = lane >> 4;          // 0: K=0,1  1: K=2,3 (A/B layout)
    const int tileBase = blockIdx.x * 128 + wave * 16;

    int node = tileBase + row;
    if (node > N_NODES - 1) node = N_NODES - 1;   // clamp; stores predicated

    // ---- A operands (16x4 f32 layout: v.x -> K=0|2, v.y -> K=1|3) ----
    float scale = 1.0f / fmaxf(cnt[node], 1.0f);
    int k0 = hi ? 2 : 0;
    v2f a_mean, a_x;
    a_mean.x = sumx[node * 3 + k0] * scale;
    a_mean.y = hi ? 0.0f : sumx[node * 3 + 1] * scale;
    a_x.x = x[node * 3 + k0];
    a_x.y = hi ? 0.0f : x[node * 3 + 1];

    // ---- B operands (4x16 f32, K rows; K=3 row is zero pad) ----
    v2f bl0, bl1, br0, br1;
    bl0.x = W1l[k0 * 32 + row];
    bl0.y = hi ? 0.0f : W1l[32 + row];
    bl1.x = W1l[k0 * 32 + 16 + row];
    bl1.y = hi ? 0.0f : W1l[32 + 16 + row];
    br0.x = W1r[k0 * 32 + row];
    br0.y = hi ? 0.0f : W1r[32 + row];
    br1.x = W1r[k0 * 32 + 16 + row];
    br1.y = hi ? 0.0f : W1r[32 + 16 + row];

    // ---- C preloaded with bias (bias depends only on the N column) ----
    float bias0 = b1[row], bias1 = b1[16 + row];
    v8f acc0, acc1;
#pragma unroll
    for (int i = 0; i < 8; ++i) { acc0[i] = bias0; acc1[i] = bias1; }

    // ---- 4x V_WMMA_F32_16X16X4_F32 : D = mean@W1l + x@W1r + b1 ----
    acc0 = __builtin_amdgcn_wmma_f32_16x16x4_f32(false, a_mean, false, bl0,
                                                 (short)0, acc0, false, false);
    acc0 = __builtin_amdgcn_wmma_f32_16x16x4_f32(false, a_x, false, br0,
                                                 (short)0, acc0, false, false);
    acc1 = __builtin_amdgcn_wmma_f32_16x16x4_f32(false, a_mean, false, bl1,
                                                 (short)0, acc1, false, false);
    acc1 = __builtin_amdgcn_wmma_f32_16x16x4_f32(false, a_x, false, br1,
                                                 (short)0, acc1, false, false);

    // ---- ReLU on accumulators, spill to LDS in per-node row layout ----
    // C layout: VGPR i, lanes 0-15 -> M=i, lanes 16-31 -> M=i+8, N = row.
#pragma unroll
    for (int i = 0; i < 8; ++i) {
        int m = i + 8 * hi;
        float* dstrow = &h1s[wave * 528 + m * 33];
        dstrow[row]      = fmaxf(acc0[i], 0.0f);   // features 0..15
        dstrow[16 + row] = fmaxf(acc1[i], 0.0f);   // features 16..31
    }
    __syncthreads();

    // ---- Project h1[32] -> 2 with A2 (lanes 0-15) / B2 (lanes 16-31) ----
    const float* Wr = hi ? B2 : A2;
    const float* hrow = &h1s[wave * 528 + row * 33];
    float o0 = 0.0f, o1 = 0.0f;
#pragma unroll
    for (int n = 0; n < 32; ++n) {
        float h = hrow[n];
        o0 = fmaf(h, Wr[2 * n + 0], o0);
        o1 = fmaf(h, Wr[2 * n + 1], o1);
    }
    int outNode = tileBase + row;
    if (outNode < N_NODES) {
        float* obuf = hi ? q : p;
        obuf[outNode * 2 + 0] = o0;
        obuf[outNode * 2 + 1] = o1;
    }
}

// Edge pass 2: aggregate folded features p (only 2 floats/edge thanks to the
// head-folding identity).  p (1.6MB) and sum_p (1.6MB) stay in L2.
__global__ void sage_edge2_kernel(const int* __restrict__ ei,
                                  const float* __restrict__ p,
                                  float* __restrict__ sump) {
    int e = blockIdx.x * blockDim.x + threadIdx.x;
    if (e >= N_EDGES) return;
    int src = ei[e];
    int dst = ei[N_EDGES + e];
    atomicAdd(&sump[dst * 2 + 0], p[src * 2 + 0]);
    atomicAdd(&sump[dst * 2 + 1], p[src * 2 + 1]);
}

__global__ void sage_final_kernel(const float* __restrict__ cnt,
                                  const float* __restrict__ sump,
                                  const float* __restrict__ q,
                                  const float* __restrict__ c2,
                                  float* __restrict__ out) {
    int n = blockIdx.x * blockDim.x + threadIdx.x;
    if (n >= N_NODES) return;
    float inv = 1.0f / fmaxf(cnt[n], 1.0f);
    out[n * 2 + 0] = sump[n * 2 + 0] * inv + q[n * 2 + 0] + c2[0];
    out[n * 2 + 1] = sump[n * 2 + 1] * inv + q[n * 2 + 1] + c2[1];
}

extern "C" void kernel_launch(void* const* d_in, const int* in_sizes, int n_in,
                              void* d_out, int out_size, void* d_ws, size_t ws_size,
                              hipStream_t stream) {
    const float* x    = (const float*)d_in[0];
    const int*   ei   = (const int*)d_in[1];   // [2, E] int32
    const float* W1l  = (const float*)d_in[2];
    const float* W1r  = (const float*)d_in[3];
    const float* b1   = (const float*)d_in[4];
    const float* W2l  = (const float*)d_in[5];
    const float* W2r  = (const float*)d_in[6];
    const float* b2   = (const float*)d_in[7];
    const float* Wlin = (const float*)d_in[8];
    const float* blin = (const float*)d_in[9];
    float* out = (float*)d_out;

    float* ws   = (float*)d_ws;
    float* cnt  = ws;
    float* sumx = ws + (size_t)N_NODES;
    float* sump = ws + (size_t)4 * N_NODES;
    float* p    = ws + (size_t)6 * N_NODES;
    float* q    = ws + (size_t)8 * N_NODES;
    float* A2   = ws + (size_t)10 * N_NODES;
    float* B2   = A2 + 64;
    float* c2   = B2 + 64;

    // 1) zero accumulation buffers (cnt, sum_x, sum_p are contiguous)
    {
        int count = 6 * N_NODES;
        sage_zero_kernel<<<(count + 255) / 256, 256, 0, stream>>>(ws, count);
    }
    // 2) fold the linear head into layer-2 weights
    sage_fold_kernel<<<1, 128, 0, stream>>>(W2l, W2r, Wlin, b2, blin, A2, B2, c2);
    // 3) edge pass 1: degree + neighbor x sums
    sage_edge1_kernel<<<(N_EDGES + 255) / 256, 256, 0, stream>>>(ei, x, cnt, sumx);
    // 4) node phase: WMMA layer-1 + ReLU + folded projections -> p, q
    sage_node_wmma_kernel<<<(N_NODES + 127) / 128, 256, 0, stream>>>(
        x, cnt, sumx, W1l, W1r, b1, A2, B2, p, q);
    // 5) edge pass 2: aggregate p
    sage_edge2_kernel<<<(N_EDGES + 255) / 256, 256, 0, stream>>>(ei, p, sump);
    // 6) combine: out = mean(p) + q + c2
    sage_final_kernel<<<(N_NODES + 255) / 256, 256, 0, stream>>>(cnt, sump, q, c2, out);
}